// VGAE_49289044689461
// MI455X (gfx1250) — compile-verified
//
#include <hip/hip_runtime.h>

// VGAE pipeline for MI455X (gfx1250, wave32).
// GEMMs: V_WMMA_F32_16X16X4_F32 from double-buffered LDS tiles.
//  - layer GEMMs: GLOBAL_LOAD_ASYNC_TO_LDS_B128/B32 (transpose-on-the-fly for W)
//  - logits GEMM: Tensor Data Mover (tensor_load_to_lds + s_wait_tensorcnt),
//    hardware row padding via pad_enable to the bank-safe 36-float stride.
// The N x N logits matrix is never materialized: BCE loss fused in epilogue.

typedef __attribute__((ext_vector_type(2))) float v2f;
typedef __attribute__((ext_vector_type(8))) float v8f;

#define TPB 256           // 8 waves of 32
#define BM 128            // 8 waves x 16 rows
#define BN 64             // 4 x 16-col WMMA tiles per wave
#define KC 32             // K-chunk staged per double-buffer step
#define KCP 36            // padded LDS row stride (floats): 16B-aligned, bank-safe

// ---- CDNA5 async copy: global -> LDS, 16B per lane ----------------------
// Copies nrows x (ncols4*4) floats from row-major global (leading dim ld)
// into LDS rows of stride_f floats. nrows*ncols4 must be a multiple of TPB.
__device__ __forceinline__ void stage_tile_async(float* lds_dst, int stride_f,
                                                 const float* gsrc, int ld,
                                                 int nrows, int ncols4) {
    uint32_t base = (uint32_t)(size_t)lds_dst;   // low 32 bits = LDS byte offset
    int total = nrows * ncols4;
    for (int c = threadIdx.x; c < total; c += TPB) {
        int r  = c / ncols4;
        int cc = (c - r * ncols4) << 2;
        uint32_t loff = base + (uint32_t)(r * stride_f + cc) * 4u;
        unsigned long ga = (unsigned long)(size_t)(gsrc + (size_t)r * ld + cc);
        asm volatile("global_load_async_to_lds_b128 %0, %1, off"
                     :: "v"(loff), "v"(ga) : "memory");
    }
}

// ---- async transpose copy: KC x BN tile of row-major B -> LDS [n][KCP] ---
// Per-lane 4B scatter (async-to-LDS allows independent LDS dest per lane).
__device__ __forceinline__ void stage_tile_async_t(float* lds_dst,
                                                   const float* gsrc, int ld) {
    uint32_t base = (uint32_t)(size_t)lds_dst;
    for (int c = threadIdx.x; c < KC * BN; c += TPB) {
        int n = c & (BN - 1);
        int k = c >> 6;   // BN == 64
        uint32_t loff = base + (uint32_t)(n * KCP + k) * 4u;
        unsigned long ga = (unsigned long)(size_t)(gsrc + (size_t)k * ld + n);
        asm volatile("global_load_async_to_lds_b32 %0, %1, off"
                     :: "v"(loff), "v"(ga) : "memory");
    }
}

__device__ __forceinline__ void wait_async_barrier() {
    asm volatile("s_wait_asynccnt 0x0" ::: "memory");
    __syncthreads();
}

// ---- Tensor Data Mover staging (6-arg builtin: clang-23 / therock-10) ----
#if __has_builtin(__builtin_amdgcn_tensor_load_to_lds)
#define HAVE_TDM 1
typedef __attribute__((ext_vector_type(4))) unsigned tdm_v4u;
typedef __attribute__((ext_vector_type(8))) int tdm_v8i;
typedef __attribute__((ext_vector_type(4))) int tdm_v4i;

// Load nrows x KC fp32 tile (row-major, leading dim ld elements) into LDS
// rows padded to KCP floats via D# pad fields (interval 32 DW, amount 4 DW).
__device__ __forceinline__ void tdm_load_tile(float* lds_dst, const float* gsrc,
                                              int ld, int nrows) {
    unsigned long ga = (unsigned long)(size_t)gsrc;
    tdm_v4u g0;
    g0.x = 1u;                                    // count=1: one valid descriptor
    g0.y = (uint32_t)(size_t)lds_dst;             // lds_addr (bytes)
    g0.z = (unsigned)(ga & 0xffffffffu);          // global_addr[31:0]
    g0.w = (unsigned)((ga >> 32) & 0x01ffffffu)   // global_addr[56:32]
         | (2u << 30);                            // type=2 ("image")
    tdm_v8i g1;
    g1[0] = (int)((2u << 16)      // data_size: 4 bytes
                | (1u << 20)      // pad_enable
                | (4u << 22)      // pad_interval: 32 DWORDs (one 128B row)
                | (3u << 25));    // pad_amount: 4 DWORDs -> 36-float stride
    g1[1] = (int)((unsigned)KC << 16);                       // tensor_dim0 lo16
    g1[2] = (int)((unsigned)(nrows & 0xffff) << 16);         // td0 hi | td1 lo
    g1[3] = (int)((unsigned)KC << 16);                       // td1 hi | tile_dim0
    g1[4] = (int)(nrows & 0xffff);                           // tile_dim1 (td2=0)
    g1[5] = (int)(unsigned)ld;                               // tensor_dim0_stride lo32
    g1[6] = 0;                                               // stride0 hi | stride1 lo
    g1[7] = 0;
    tdm_v4i z4 = {0, 0, 0, 0};
    tdm_v8i z8 = {0, 0, 0, 0, 0, 0, 0, 0};
    __builtin_amdgcn_tensor_load_to_lds(g0, g1, z4, z4, z8, 0);
}
#endif

__device__ __forceinline__ float block_reduce_sum(float v, float* red) {
    int tid = threadIdx.x;
    red[tid] = v;
    __syncthreads();
    for (int s = TPB / 2; s > 0; s >>= 1) {
        if (tid < s) red[tid] += red[tid + s];
        __syncthreads();
    }
    return red[0];
}

// ---- degree: cnt[dst] += 1 per edge -------------------------------------
__global__ void deg_kernel(const int* __restrict__ ei, float* __restrict__ cnt, int E) {
    int e = blockIdx.x * blockDim.x + threadIdx.x;
    if (e < E) atomicAdd(&cnt[ei[E + e]], 1.0f);
}

// ---- scatter: agg[dst] += x[src] * w, 4 floats per thread ----------------
__global__ void scatter_kernel(const float* __restrict__ x, const int* __restrict__ ei,
                               const float* __restrict__ w, float* __restrict__ agg,
                               int E, int D) {
    long idx = (long)blockIdx.x * blockDim.x + threadIdx.x;
    int vecs = D >> 2;
    long total = (long)E * vecs;
    if (idx >= total) return;
    int e = (int)(idx / vecs);
    int d = ((int)(idx % vecs)) << 2;
    int s = ei[e], t = ei[E + e];
    float we = w[e];
    const float4 xv = *(const float4*)(x + (long)s * D + d);
    float* ap = agg + (long)t * D + d;
    atomicAdd(ap + 0, xv.x * we);
    atomicAdd(ap + 1, xv.y * we);
    atomicAdd(ap + 2, xv.z * we);
    atomicAdd(ap + 3, xv.w * we);
}

// ---- hcat = [x, agg / max(cnt,1)] ---------------------------------------
__global__ void concat_kernel(const float* __restrict__ x, const float* __restrict__ agg,
                              const float* __restrict__ cnt, float* __restrict__ hcat,
                              int Nn, int D) {
    int idx = blockIdx.x * blockDim.x + threadIdx.x;
    if (idx >= Nn * D) return;
    int n = idx / D, d = idx % D;
    float inv = 1.0f / fmaxf(cnt[n], 1.0f);
    hcat[(long)n * 2 * D + d]     = x[idx];
    hcat[(long)n * 2 * D + D + d] = agg[idx] * inv;
}

// ---- fp32 WMMA GEMM: out = act(A[M,K] @ B[K,N] + bias) (+ residual) ------
// A row-major [M,K] lda; B row-major [K,N] ldb. M%128==0, N%64==0, K%32==0.
__global__ __launch_bounds__(TPB) void gemm_wmma_f32(
    const float* __restrict__ A, int lda,
    const float* __restrict__ B, int ldb,
    float* __restrict__ Cout, int ldc,
    int M, int Nc, int Kc,
    const float* __restrict__ bias, int do_relu,
    const float* __restrict__ residual) {
    __shared__ float sA[2][BM * KCP];   // A tile: [row][k], padded
    __shared__ float sB[2][BN * KCP];   // B tile transposed: [n][k], padded

    int wave = threadIdx.x >> 5;
    int lane = threadIdx.x & 31;
    int m0 = blockIdx.x * BM;
    int n0 = blockIdx.y * BN;
    int l15 = lane & 15;
    int khalf = (lane >> 4) << 1;       // 0 or 2: K sub-pair for this lane half
    int arow = (wave << 4) + l15;       // 0..127

    // prologue: stage chunk 0
    stage_tile_async(sA[0], KCP, A + (size_t)m0 * lda, lda, BM, KC / 4);
    stage_tile_async_t(sB[0], B + n0, ldb);
    wait_async_barrier();

    v8f acc[4] = {};
    int nch = Kc / KC;
    for (int c = 0; c < nch; ++c) {
        int p = c & 1;
        if (c + 1 < nch) {
            int k1 = (c + 1) * KC;
            stage_tile_async(sA[1 - p], KCP, A + (size_t)m0 * lda + k1, lda, BM, KC / 4);
            stage_tile_async_t(sB[1 - p], B + (size_t)k1 * ldb + n0, ldb);
        }
        const float* __restrict__ pa = &sA[p][arow * KCP];
        const float* __restrict__ pb = &sB[p][0];
#pragma unroll
        for (int kk = 0; kk < KC; kk += 4) {
            v2f a;
            a.x = pa[kk + khalf];
            a.y = pa[kk + khalf + 1];
#pragma unroll
            for (int t = 0; t < 4; ++t) {
                const float* bp = pb + ((t << 4) + l15) * KCP + kk + khalf;
                v2f b;
                b.x = bp[0];
                b.y = bp[1];
                acc[t] = __builtin_amdgcn_wmma_f32_16x16x4_f32(
                    false, a, false, b, (short)0, acc[t], false, false);
            }
        }
        wait_async_barrier();   // next chunk resident before reuse
    }

    int rbase = (lane >> 4) << 3;       // lanes 16-31 hold M = r + 8
#pragma unroll
    for (int t = 0; t < 4; ++t) {
        int col = n0 + (t << 4) + l15;
        float bv = bias ? bias[col] : 0.0f;
#pragma unroll
        for (int r = 0; r < 8; ++r) {
            int row = m0 + (wave << 4) + rbase + r;
            float v = acc[t][r] + bv;
            if (do_relu) v = fmaxf(v, 0.0f);
            if (residual) v += residual[(long)row * ldc + col];
            Cout[(long)row * ldc + col] = v;
        }
    }
}

// ---- fused logits GEMM + BCEWithLogits reduction -------------------------
// logits = Zl @ Zl^T (register tiles only); loss += softplus(x) - x*ADJ.
// Tiles staged by the Tensor Data Mover (one descriptor per tile per chunk).
__global__ __launch_bounds__(TPB) void gemm_loss_wmma(
    const float* __restrict__ Zl, const float* __restrict__ ADJ,
    float* __restrict__ acc_loss, int Nn, int Kc) {
    __shared__ float sA[2][BM * KCP];   // rows m0..m0+127 of Zl
    __shared__ float sB[2][BN * KCP];   // rows n0..n0+63 of Zl (= Zl^T cols)
    __shared__ float red[TPB];

    int wave = threadIdx.x >> 5;
    int lane = threadIdx.x & 31;
    int m0 = blockIdx.x * BM;
    int n0 = blockIdx.y * BN;
    int l15 = lane & 15;
    int khalf = (lane >> 4) << 1;
    int arow = (wave << 4) + l15;
    bool issuer = (threadIdx.x < 32);   // TDM ignores EXEC: issue from wave 0 only

#if HAVE_TDM
    if (issuer) {
        tdm_load_tile(sA[0], Zl + (size_t)m0 * Kc, Kc, BM);
        tdm_load_tile(sB[0], Zl + (size_t)n0 * Kc, Kc, BN);
        __builtin_amdgcn_s_wait_tensorcnt(0);
    }
    __syncthreads();
#else
    stage_tile_async(sA[0], KCP, Zl + (size_t)m0 * Kc, Kc, BM, KC / 4);
    stage_tile_async(sB[0], KCP, Zl + (size_t)n0 * Kc, Kc, BN, KC / 4);
    wait_async_barrier();
#endif

    v8f acc[4] = {};
    int nch = Kc / KC;
    for (int c = 0; c < nch; ++c) {
        int p = c & 1;
        if (c + 1 < nch) {
            int k1 = (c + 1) * KC;
#if HAVE_TDM
            if (issuer) {
                tdm_load_tile(sA[1 - p], Zl + (size_t)m0 * Kc + k1, Kc, BM);
                tdm_load_tile(sB[1 - p], Zl + (size_t)n0 * Kc + k1, Kc, BN);
            }
#else
            stage_tile_async(sA[1 - p], KCP, Zl + (size_t)m0 * Kc + k1, Kc, BM, KC / 4);
            stage_tile_async(sB[1 - p], KCP, Zl + (size_t)n0 * Kc + k1, Kc, BN, KC / 4);
#endif
        }
        const float* __restrict__ pa = &sA[p][arow * KCP];
        const float* __restrict__ pb = &sB[p][0];
#pragma unroll
        for (int kk = 0; kk < KC; kk += 4) {
            v2f a;
            a.x = pa[kk + khalf];
            a.y = pa[kk + khalf + 1];
#pragma unroll
            for (int t = 0; t < 4; ++t) {
                const float* bp = pb + ((t << 4) + l15) * KCP + kk + khalf;
                v2f b;
                b.x = bp[0];
                b.y = bp[1];
                acc[t] = __builtin_amdgcn_wmma_f32_16x16x4_f32(
                    false, a, false, b, (short)0, acc[t], false, false);
            }
        }
#if HAVE_TDM
        if (issuer) __builtin_amdgcn_s_wait_tensorcnt(0);
        __syncthreads();
#else
        wait_async_barrier();
#endif
    }

    int rbase = (lane >> 4) << 3;
    float local = 0.0f;
#pragma unroll
    for (int t = 0; t < 4; ++t) {
        int col = n0 + (t << 4) + l15;
#pragma unroll
        for (int r = 0; r < 8; ++r) {
            int row = m0 + (wave << 4) + rbase + r;
            float x = acc[t][r];
            float sp = fmaxf(x, 0.0f) + log1pf(expf(-fabsf(x)));  // stable softplus
            local += sp - x * ADJ[(long)row * Nn + col];
        }
    }
    float s = block_reduce_sum(local, red);
    if (threadIdx.x == 0) atomicAdd(acc_loss, s);
}

// ---- Z = mu + eps*exp(.5*lv); Z_l = mean_k(Z); KL partial sums -----------
__global__ void z_kl_kernel(const float* __restrict__ mu, const float* __restrict__ lv,
                            const float* __restrict__ eps, float* __restrict__ Zout,
                            float* __restrict__ Zl, float* __restrict__ kl_acc,
                            int total, int D, int K) {
    __shared__ float red[TPB];
    int idx = blockIdx.x * blockDim.x + threadIdx.x;
    float local = 0.0f;
    if (idx < total) {
        int n = idx / D, d = idx % D;
        float m = mu[idx], l = lv[idx];
        float el  = expf(l);
        float sig = expf(0.5f * l);
        float es = 0.0f;
        for (int k = 0; k < K; ++k) {
            long zi = ((long)n * K + k) * D + d;
            float e = eps[zi];
            es += e;
            Zout[zi] = m + e * sig;
        }
        Zl[idx] = m + sig * (es * (1.0f / K));
        local = 1.0f + l - m * m - el;
    }
    float s = block_reduce_sum(local, red);
    if (threadIdx.x == 0) atomicAdd(kl_acc, s);
}

__global__ void finalize_kernel(const float* __restrict__ acc, float* __restrict__ out,
                                float invN, float invNN) {
    out[0] = -0.5f * acc[0] * invN + acc[1] * invNN;
}

extern "C" void kernel_launch(void* const* d_in, const int* in_sizes, int n_in,
                              void* d_out, int out_size, void* d_ws, size_t ws_size,
                              hipStream_t stream) {
    (void)n_in; (void)out_size; (void)ws_size;
    const int N = 8192, D = 256, K = 8;
    const float* enc = (const float*)d_in[0];
    const int*   ei  = (const int*)d_in[1];
    const float* w   = (const float*)d_in[2];
    const float* ADJ = (const float*)d_in[3];
    const float* eps = (const float*)d_in[4];
    const float* W1  = (const float*)d_in[5];
    const float* b1  = (const float*)d_in[6];
    const float* W2  = (const float*)d_in[7];
    const float* b2  = (const float*)d_in[8];
    const float* W3  = (const float*)d_in[9];
    const float* b3  = (const float*)d_in[10];
    const float* W4  = (const float*)d_in[11];
    const float* b4  = (const float*)d_in[12];
    const int E = in_sizes[2];

    // workspace layout (floats)
    float* ws   = (float*)d_ws;
    float* cnt  = ws;                        // N
    float* agg  = cnt + N;                   // N*D
    float* hcat = agg + (size_t)N * D;       // N*2D (reused for both conv layers)
    float* h1   = hcat + (size_t)N * 2 * D;  // N*D (reused as Z_l afterwards)
    float* mu   = h1 + (size_t)N * D;        // N*D
    float* lv   = mu + (size_t)N * D;        // N*D
    float* acc2 = lv + (size_t)N * D;        // [kl_sum, loss_sum]
    float* Zl   = h1;

    float* enc2 = (float*)d_out;
    float* Z    = enc2 + (size_t)N * D;
    float* loss = Z + (size_t)N * K * D;

    (void)hipMemsetAsync(cnt, 0, (size_t)(N + (size_t)N * D) * sizeof(float), stream);
    (void)hipMemsetAsync(acc2, 0, 2 * sizeof(float), stream);

    // layer 1 conv
    deg_kernel<<<(E + TPB - 1) / TPB, TPB, 0, stream>>>(ei, cnt, E);
    {
        long tot = (long)E * (D / 4);
        scatter_kernel<<<(tot + TPB - 1) / TPB, TPB, 0, stream>>>(enc, ei, w, agg, E, D);
    }
    concat_kernel<<<(N * D + TPB - 1) / TPB, TPB, 0, stream>>>(enc, agg, cnt, hcat, N, D);
    gemm_wmma_f32<<<dim3(N / BM, D / BN), TPB, 0, stream>>>(
        hcat, 2 * D, W1, D, h1, D, N, D, 2 * D, b1, 1, nullptr);

    // layer 2 conv
    (void)hipMemsetAsync(agg, 0, (size_t)N * D * sizeof(float), stream);
    {
        long tot = (long)E * (D / 4);
        scatter_kernel<<<(tot + TPB - 1) / TPB, TPB, 0, stream>>>(h1, ei, w, agg, E, D);
    }
    concat_kernel<<<(N * D + TPB - 1) / TPB, TPB, 0, stream>>>(h1, agg, cnt, hcat, N, D);
    gemm_wmma_f32<<<dim3(N / BM, D / BN), TPB, 0, stream>>>(
        hcat, 2 * D, W2, D, enc2, D, N, D, 2 * D, b2, 1, enc);  // enc2 = relu(.)+enc

    // VAE heads
    gemm_wmma_f32<<<dim3(N / BM, D / BN), TPB, 0, stream>>>(
        enc2, D, W3, D, mu, D, N, D, D, b3, 0, nullptr);
    gemm_wmma_f32<<<dim3(N / BM, D / BN), TPB, 0, stream>>>(
        enc2, D, W4, D, lv, D, N, D, D, b4, 0, nullptr);

    // reparameterize + KL
    z_kl_kernel<<<(N * D + TPB - 1) / TPB, TPB, 0, stream>>>(
        mu, lv, eps, Z, Zl, acc2, N * D, D, K);

    // fused Z_l @ Z_l^T + BCEWithLogits reduction (logits never hit HBM)
    gemm_loss_wmma<<<dim3(N / BM, N / BN), TPB, 0, stream>>>(Zl, ADJ, acc2 + 1, N, D);

    finalize_kernel<<<1, 1, 0, stream>>>(acc2, loss, 1.0f / N, 1.0f / ((float)N * (float)N));
}